// QueryAndGroup_6390911336633
// MI455X (gfx1250) — compile-verified
//
#include <hip/hip_runtime.h>

// Problem constants (match reference)
#define B_      4
#define N_      16384
#define S_      4096
#define C_      64
#define K_      32          // NSAMPLE
#define R2_     0.04f       // RADIUS^2
#define CHUNK   2048        // points staged in LDS per iteration (24 KB raw)
#define WAVES   8           // waves per workgroup
#define CPW     (WAVES*16)  // centroids per workgroup (each wave owns 16)

typedef __attribute__((ext_vector_type(2))) float v2f;
typedef __attribute__((ext_vector_type(8))) float v8f;
typedef __attribute__((ext_vector_type(4))) int   v4i;

typedef __attribute__((address_space(1))) v4i gas_v4i;   // global
typedef __attribute__((address_space(3))) v4i las_v4i;   // LDS

#if defined(__has_builtin)
#if __has_builtin(__builtin_amdgcn_global_load_async_to_lds_b128)
#define HAVE_ASYNC_LDS 1
#endif
#endif

#if defined(__AMDGCN__) && !defined(HAVE_ASYNC_LDS)
#warning "device pass: global_load_async_to_lds builtin NOT available, using sync fallback"
#endif

__device__ __forceinline__ unsigned ballot32(bool p) {
    return __builtin_amdgcn_ballot_w32(p);
}

__device__ __forceinline__ void wait_async0() {
#if defined(__has_builtin) && __has_builtin(__builtin_amdgcn_s_wait_asynccnt)
    __builtin_amdgcn_s_wait_asynccnt(0);
#else
    asm volatile("s_wait_asynccnt 0x0" ::: "memory");
#endif
}

// ---------------------------------------------------------------------------
// Phase 1: ball query via V_WMMA_F32_16X16X4_F32 + async global->LDS staging.
//   WMMA computes  d[m][n] = |q_m|^2 - 2 q_m.p_n   (C preloaded with |q|^2)
//     A (16x4): lane<16 -> {-2qx,-2qy}, lane>=16 -> {-2qz, 0}
//     B (4x16): raw xyz from LDS; row K3 is garbage (A[K3]=0 cancels it)
//   In-ball test: d[r] < R^2 - |p_n|^2; threshold computed per lane/column
//   from 3 conflict-free b32 LDS reads (no cross-lane exchange needed).
// Staging: raw 12B/point copied with GLOBAL_LOAD_ASYNC_TO_LDS_B128 (ASYNCcnt),
// 6 x b128 per thread per 24KB chunk.
// Selection: ballot per accumulator VGPR; lanes 0-15 own one row each and pop
// bits in ascending index order until 32 found; wave + workgroup early exit.
// ---------------------------------------------------------------------------
__global__ __launch_bounds__(256)
void ball_query_wmma(const float* __restrict__ xyz,
                     const float* __restrict__ nxyz,
                     int* __restrict__ idx) {
    __shared__ float pts3[CHUNK * 3];     // raw {x,y,z} per point, 24 KB
    __shared__ int done_ctr;

    const int tid  = threadIdx.x;
    const int wave = tid >> 5;
    const int lane = tid & 31;
    const int wgPerB = S_ / CPW;                       // 32
    const int b    = blockIdx.x / wgPerB;
    const int sw   = (blockIdx.x % wgPerB) * CPW + wave * 16;
    const int m    = lane & 15;
    const int srow = sw + m;
    const bool lo  = (lane < 16);

    // centroid for row m of this wave's tile
    const float* q = nxyz + ((size_t)b * S_ + srow) * 3;
    const float qx = q[0], qy = q[1], qz = q[2];
    const float q2 = qx * qx + qy * qy + qz * qz;

    v2f a;
    if (lo) { a[0] = -2.0f * qx; a[1] = -2.0f * qy; }  // K0,K1
    else    { a[0] = -2.0f * qz; a[1] = 0.0f;       }  // K2,K3 (K3=0)

    // accumulator preload: qn[r] = |q_{r + 8*hi}|^2  (C/D VGPR layout)
    const int hi = lo ? 0 : 8;
    v8f qn;
#pragma unroll
    for (int r = 0; r < 8; ++r) qn[r] = __shfl(q2, hi + r, 32);

    if (tid == 0) done_ctr = 0;

    int cnt   = lo ? 0 : K_;            // lanes 16-31 dummy-full for ballots
    int first = 0;
    const size_t ob = ((size_t)b * S_ + srow) * K_;
    bool wdone = false;

    __syncthreads();

    for (int cb = 0; cb < N_; cb += CHUNK) {
        if (*(volatile int*)&done_ctr == WAVES) break;   // uniform across WG

        // ---- stage raw chunk: CHUNK*12 = 24576 B = 1536 x 16 B ----
        const char* gsrc = (const char*)(xyz + ((size_t)b * N_ + cb) * 3);
#if defined(HAVE_ASYNC_LDS)
        {
            gas_v4i* g = (gas_v4i*)gsrc;                 // 16B elements
            las_v4i* l = (las_v4i*)pts3;
#pragma unroll
            for (int i = 0; i < 6; ++i) {
                __builtin_amdgcn_global_load_async_to_lds_b128(
                    g + tid + i * 256,
                    l + tid + i * 256, 0, 0);
            }
            wait_async0();
        }
#else
        {
            const float* gf = (const float*)gsrc;
            for (int t = tid; t < CHUNK * 3; t += 256) pts3[t] = gf[t];
        }
#endif
        __syncthreads();

        if (!wdone) {
            for (int t0 = 0; t0 < CHUNK; t0 += 16) {
                const int b3 = (t0 + m) * 3;
                // 3 conflict-free b32 reads (lanes l / l+16 share b3+1 ->
                // LDS broadcast). lo: {px,py,pz}; hi: {pz,py,px}.
                const float e0 = pts3[b3 + (lo ? 0 : 2)];
                const float e1 = pts3[b3 + 1];
                const float e2 = pts3[b3 + (lo ? 2 : 0)];
                v2f bv; bv[0] = e0; bv[1] = e1;   // K3 row garbage is harmless

                // per-column |p|^2 -> threshold (no per-row adds, no shuffle)
                const float p2  = e0 * e0 + e1 * e1 + e2 * e2;
                const float thr = R2_ - p2;

                v8f d = __builtin_amdgcn_wmma_f32_16x16x4_f32(
                    false, a, false, bv, (short)0, qn, false, false);

                const int nb = cb + t0;
#pragma unroll
                for (int r = 0; r < 8; ++r) {
                    const unsigned msk = ballot32(d[r] < thr);
                    unsigned my = (((m & 7) == r) && lo)
                        ? ((msk >> ((m >> 3) * 16)) & 0xFFFFu) : 0u;
                    while (my && cnt < K_) {
                        const int bit  = __builtin_ctz(my);
                        const int nidx = nb + bit;
                        if (cnt == 0) first = nidx;
                        idx[ob + cnt] = nidx;
                        ++cnt;
                        my &= my - 1;
                    }
                }
                if (ballot32(cnt >= K_) == 0xFFFFFFFFu) break;  // wave-uniform
            }
            if (ballot32(cnt >= K_) == 0xFFFFFFFFu) {
                wdone = true;
                if (lane == 0) atomicAdd(&done_ctr, 1);
            }
        }
        __syncthreads();
    }

    // pad empty slots with first found index (0 if none) — reference semantics
    if (lo) {
        for (int k = cnt; k < K_; ++k) idx[ob + k] = first;
    }
}

// ---------------------------------------------------------------------------
// Phase 2: gather + concat. One wave per centroid; lane = sample slot, so
// each store of 32 lanes is one contiguous 128B line. Feature gathers hit L2
// (16 MB resident vs 192 MB L2). Prefetch upcoming feature rows.
// out[b][c][s][k]: c<3 -> xyz[idx]-new_xyz ; else features[c-3][idx]
// ---------------------------------------------------------------------------
__global__ __launch_bounds__(256)
void group_gather(const float* __restrict__ xyz,
                  const float* __restrict__ nxyz,
                  const float* __restrict__ feat,
                  const int* __restrict__ idx,
                  float* __restrict__ out) {
    const int gw   = blockIdx.x * 8 + (threadIdx.x >> 5);
    const int lane = threadIdx.x & 31;
    const int b = gw / S_;
    const int s = gw % S_;

    const int j = idx[((size_t)b * S_ + s) * K_ + lane];

    const float* p = xyz + ((size_t)b * N_ + j) * 3;
    const float px = p[0], py = p[1], pz = p[2];
    const float* q = nxyz + ((size_t)b * S_ + s) * 3;

    const size_t cs = (size_t)S_ * K_;                 // channel stride
    const size_t o  = ((size_t)b * (C_ + 3)) * cs + (size_t)s * K_ + lane;

    out[o]          = px - q[0];
    out[o + cs]     = py - q[1];
    out[o + 2 * cs] = pz - q[2];

    const float* f  = feat + ((size_t)b * C_) * N_ + j;
    float* og = out + o + 3 * cs;
#pragma unroll 4
    for (int c = 0; c < C_; ++c) {
        if (c + 8 < C_) __builtin_prefetch(f + (size_t)(c + 8) * N_, 0, 0);
        og[(size_t)c * cs] = f[(size_t)c * N_];
    }
}

// ---------------------------------------------------------------------------
extern "C" void kernel_launch(void* const* d_in, const int* in_sizes, int n_in,
                              void* d_out, int out_size, void* d_ws, size_t ws_size,
                              hipStream_t stream) {
    const float* xyz  = (const float*)d_in[0];   // [B,N,3]
    const float* nxyz = (const float*)d_in[1];   // [B,S,3]
    const float* feat = (const float*)d_in[2];   // [B,C,N]
    float* out = (float*)d_out;                  // [B,C+3,S,K]
    int* idxws = (int*)d_ws;                     // [B,S,K] int32 (2 MB)

    (void)in_sizes; (void)n_in; (void)out_size; (void)ws_size;

    dim3 g1(B_ * (S_ / CPW));                    // 128 workgroups
    ball_query_wmma<<<g1, 256, 0, stream>>>(xyz, nxyz, idxws);

    dim3 g2((B_ * S_) / 8);                      // 2048 workgroups
    group_gather<<<g2, 256, 0, stream>>>(xyz, nxyz, feat, idxws, out);
}